// TensorizedAutoencoder_87900800680230
// MI455X (gfx1250) — compile-verified
//
#include <hip/hip_runtime.h>

#define B_N 2048
#define D_N 512
#define H_N 128
#define K_N 16
#define TPC (B_N / 16)   // max 16-row tiles per cluster = 128

typedef __attribute__((ext_vector_type(2))) float v2f;
typedef __attribute__((ext_vector_type(8))) float v8f;

// ---------------- Kernel 1: nearest-center assignment + histogram ----------------
__global__ __launch_bounds__(256) void assign_kernel(
    const float* __restrict__ x, const float* __restrict__ centers,
    int* __restrict__ idx, int* __restrict__ counts)
{
    __shared__ float c[K_N][D_N];   // 32 KB
    for (int i = threadIdx.x; i < K_N * D_N; i += 256)
        c[i / D_N][i % D_N] = centers[i];
    __syncthreads();

    int b = blockIdx.x * 256 + threadIdx.x;
    if (b >= B_N) return;
    const float* xb = x + (size_t)b * D_N;

    float dist[K_N];
#pragma unroll
    for (int k = 0; k < K_N; ++k) dist[k] = 0.f;

    for (int d = 0; d < D_N; ++d) {
        float xv = xb[d];
#pragma unroll
        for (int k = 0; k < K_N; ++k) {
            float t = xv - c[k][d];          // LDS read is a broadcast (same addr per wave)
            dist[k] = fmaf(t, t, dist[k]);
        }
    }
    int best = 0; float bd = dist[0];
#pragma unroll
    for (int k = 1; k < K_N; ++k) if (dist[k] < bd) { bd = dist[k]; best = k; }
    idx[b] = best;
    atomicAdd(&counts[best], 1);
}

// ---------------- Kernel 2: prefix-sum offsets + compaction into cluster order ----
__global__ __launch_bounds__(256) void scan_scatter_kernel(
    const int* __restrict__ idx, const int* __restrict__ counts,
    int* __restrict__ offsets, int* __restrict__ cursor, int* __restrict__ order)
{
    __shared__ int soff[K_N];
    if (threadIdx.x == 0) {
        int acc = 0;
        for (int k = 0; k < K_N; ++k) { soff[k] = acc; offsets[k] = acc; acc += counts[k]; }
    }
    __syncthreads();
    for (int b = threadIdx.x; b < B_N; b += 256) {
        int k = idx[b];
        int pos = soff[k] + atomicAdd(&cursor[k], 1);
        order[pos] = b;
    }
}

// ---------------- Kernel 3: fused per-cluster encoder+decoder (WMMA fp32) --------
// block = (cluster k, tile t): 16 rows of cluster k. 256 threads = 8 waves (wave32).
__global__ __launch_bounds__(256) void ae_kernel(
    const float* __restrict__ x, const float* __restrict__ centers,
    const float* __restrict__ encW, const float* __restrict__ encB,
    const float* __restrict__ decW, const float* __restrict__ decB,
    const int* __restrict__ counts, const int* __restrict__ offsets,
    const int* __restrict__ order, float* __restrict__ out)
{
    __shared__ float sXC[16][D_N];  // 32 KB: center-subtracted inputs
    __shared__ float sE[16][H_N];   //  8 KB: hidden activations
    __shared__ int   sRows[16];

    const int k = blockIdx.x / TPC;
    const int t = blockIdx.x % TPC;
    const int cnt = counts[k];
    if (t * 16 >= cnt) return;                    // uniform exit: EXEC all-1s below
    const int base   = offsets[k] + t * 16;
    const int nvalid = min(16, cnt - t * 16);

    const int tid = threadIdx.x;
    if (tid < 16) sRows[tid] = (tid < nvalid) ? order[base + tid] : -1;
    __syncthreads();

    const float* ck = centers + (size_t)k * D_N;
    for (int i = tid; i < 16 * D_N; i += 256) {
        int m = i >> 9;            // i / 512
        int d = i & (D_N - 1);
        int r = sRows[m];
        sXC[m][d] = (r >= 0) ? (x[(size_t)r * D_N + d] - ck[d]) : 0.f;
    }
    __syncthreads();

    const int lane = tid & 31;
    const int wv   = tid >> 5;      // wave id 0..7
    const int ll   = lane & 15;     // row (A/C) or column (B/C) index
    const int hi   = lane >> 4;     // lane-half selector

    // ---- encoder: E = relu(XC[16,512] @ encW_k[512,128] + encB_k)
    // wave wv owns output columns [16*wv, 16*wv+16)
    const float* Wk = encW + (size_t)k * D_N * H_N;
    const int n = wv * 16 + ll;
    v8f acc = {};
    for (int kb = 0; kb < D_N; kb += 4) {
        // A 16x4 layout: VGPR0 = K=2*hi, VGPR1 = K=2*hi+1, row = ll  -> ds_load_b64
        v2f a = *(const v2f*)&sXC[ll][kb + 2 * hi];
        // B 4x16 layout: VGPR j holds row K=j+2*hi, col = ll
        int kr = kb + 2 * hi;
        v2f bb;
        bb.x = Wk[(size_t)kr * H_N + n];
        bb.y = Wk[(size_t)(kr + 1) * H_N + n];
        acc = __builtin_amdgcn_wmma_f32_16x16x4_f32(
            false, a, false, bb, (short)0, acc, false, false);
    }
    {
        float eb = encB[k * H_N + n];
#pragma unroll
        for (int j = 0; j < 8; ++j) {
            float e = acc[j] + eb;               // C/D: row M = j + 8*hi, col = n
            e = e > 0.f ? e : 0.f;
            sE[j + 8 * hi][n] = e;
        }
    }
    __syncthreads();

    // ---- decoder: OUT = E[16,128] @ decW_k[128,512] + decB_k
    // 32 N-tiles of 16; each wave handles 4 (nt = wv, wv+8, wv+16, wv+24)
    const float* Dk = decW + (size_t)k * H_N * D_N;
    for (int nt = wv; nt < D_N / 16; nt += 8) {
        const int nn = nt * 16 + ll;
        v8f dacc = {};
        for (int kb = 0; kb < H_N; kb += 4) {
            v2f a = *(const v2f*)&sE[ll][kb + 2 * hi];
            int kr = kb + 2 * hi;
            v2f bb;
            bb.x = Dk[(size_t)kr * D_N + nn];
            bb.y = Dk[(size_t)(kr + 1) * D_N + nn];
            dacc = __builtin_amdgcn_wmma_f32_16x16x4_f32(
                false, a, false, bb, (short)0, dacc, false, false);
        }
        float db = decB[k * D_N + nn];
#pragma unroll
        for (int j = 0; j < 8; ++j) {
            int m = j + 8 * hi;                  // output row within tile
            if (m < nvalid) {
                int r = sRows[m];
                out[(size_t)r * D_N + nn] = dacc[j] + db;
            }
        }
    }
}

extern "C" void kernel_launch(void* const* d_in, const int* in_sizes, int n_in,
                              void* d_out, int out_size, void* d_ws, size_t ws_size,
                              hipStream_t stream) {
    (void)in_sizes; (void)n_in; (void)out_size; (void)ws_size;
    const float* x       = (const float*)d_in[0];
    const float* centers = (const float*)d_in[1];
    const float* encW    = (const float*)d_in[2];
    const float* encB    = (const float*)d_in[3];
    const float* decW    = (const float*)d_in[4];
    const float* decB    = (const float*)d_in[5];
    float* out = (float*)d_out;

    // workspace layout (ints)
    int* idx     = (int*)d_ws;          // [B_N]
    int* counts  = idx + B_N;           // [K_N]
    int* offsets = counts + K_N;        // [K_N]
    int* cursor  = offsets + K_N;       // [K_N]
    int* order   = cursor + K_N;        // [B_N]

    // zero counts/offsets/cursor every call (deterministic, capture-safe)
    hipMemsetAsync(counts, 0, 3 * K_N * sizeof(int), stream);

    assign_kernel<<<B_N / 256, 256, 0, stream>>>(x, centers, idx, counts);
    scan_scatter_kernel<<<1, 256, 0, stream>>>(idx, counts, offsets, cursor, order);
    ae_kernel<<<K_N * TPC, 256, 0, stream>>>(x, centers, encW, encB, decW, decB,
                                             counts, offsets, order, out);
}